// GCNFeatureExtractor_87780541596432
// MI455X (gfx1250) — compile-verified
//
#include <hip/hip_runtime.h>
#include <math.h>

typedef __attribute__((ext_vector_type(2))) float v2f;
typedef __attribute__((ext_vector_type(8))) float v8f;

#define J_N   75
#define JP    80      // row padding (5 M-tiles of 16)
#define KP    76      // K padding for the A-GEMMs (19 k-steps of 4)
#define KSTEPS 19
#define D_IN  3
#define HID   64
#define OUT_D 256
#define E_EDG 300
#define BT    2048

__device__ __forceinline__ float gelu_exact(float v) {
    return 0.5f * v * (1.0f + erff(v * 0.70710678118654752f));
}

// ---------------------------------------------------------------------------
// Kernel 1: build dense normalized adjacency Â (JP x JP, zero-padded) in ws.
//   deg[j] = 1 (self loop) + #incoming edges;  Â[d][s] += deg[s]^-1/2 * deg[d]^-1/2
// Duplicate edges accumulate, exactly matching the reference scatter-add.
// ---------------------------------------------------------------------------
__global__ void build_adj_kernel(const int* __restrict__ edge_index,
                                 float* __restrict__ A) {
    __shared__ float deg[J_N];
    __shared__ float dis[J_N];
    const int tid = threadIdx.x;

    for (int j = tid; j < J_N; j += blockDim.x) deg[j] = 1.0f;  // self loop
    __syncthreads();
    for (int i = tid; i < E_EDG; i += blockDim.x) {
        int d = edge_index[E_EDG + i];
        atomicAdd(&deg[d], 1.0f);
    }
    __syncthreads();
    for (int j = tid; j < J_N; j += blockDim.x) dis[j] = rsqrtf(deg[j]);
    for (int i = tid; i < JP * JP; i += blockDim.x) A[i] = 0.0f;
    __syncthreads();
    for (int i = tid; i < E_EDG; i += blockDim.x) {
        int s = edge_index[i];
        int d = edge_index[E_EDG + i];
        atomicAdd(&A[d * JP + s], dis[s] * dis[d]);
    }
    for (int j = tid; j < J_N; j += blockDim.x) {
        atomicAdd(&A[j * JP + j], dis[j] * dis[j]);
    }
}

// ---------------------------------------------------------------------------
// Kernel 2: fused 2-layer GCN per graph, matmuls except X·W1 (VALU) via V_WMMA_F32_16X16X4_F32
//   h1  = gelu(Â·(X·W1) + b1)
//   out = gelu((Â·h1)·W2 + b2)        (associativity: Â·(h1·W2) == (Â·h1)·W2)
// ---------------------------------------------------------------------------
__launch_bounds__(256)
__global__ void gcn_fused_kernel(const float* __restrict__ x,
                                 const float* __restrict__ W1,
                                 const float* __restrict__ b1,
                                 const float* __restrict__ W2,
                                 const float* __restrict__ b2,
                                 const float* __restrict__ Adj,
                                 float* __restrict__ out) {
    __shared__ float sA[JP * JP];       // 25.6 KB  Â (zero-padded)
    __shared__ float sH[JP * HID];      // 20.5 KB  h0 -> h1 -> G
    __shared__ float sW1[D_IN * HID];
    __shared__ float sB1[HID];
    __shared__ float sB2[OUT_D];

    const int tid   = threadIdx.x;
    const int lane  = tid & 31;
    const int wave  = tid >> 5;         // 0..7
    const int lhalf = lane >> 4;        // 0 | 1  (lane half, selects K pair)
    const int l15   = lane & 15;
    const int g     = blockIdx.x;

    // ---- stage shared data -------------------------------------------------
    for (int i = tid; i < JP * JP; i += 256) sA[i] = Adj[i];
    for (int i = tid; i < D_IN * HID; i += 256) sW1[i] = W1[i];
    for (int i = tid; i < HID; i += 256) sB1[i] = b1[i];
    for (int i = tid; i < OUT_D; i += 256) sB2[i] = b2[i];

    // ---- h0 = X · W1 (K=3, VALU), rows >= 75 zeroed ------------------------
    const float* xg = x + (size_t)g * (J_N * D_IN);
    for (int i = tid; i < JP * HID; i += 256) {
        int j = i / HID, k = i - j * HID;
        float acc = 0.0f;
        if (j < J_N) {
            acc = xg[j * 3 + 0] * sW1[0 * HID + k]
                + xg[j * 3 + 1] * sW1[1 * HID + k]
                + xg[j * 3 + 2] * sW1[2 * HID + k];
        }
        sH[i] = acc;
    }
    __syncthreads();

    // Tile assignment for the two Â-GEMMs (5x4 = 20 tiles of 16x16):
    //   wave w owns column block tn = w & 3 (constant!), rows tm = (w>>2) + 2*i.
    // B-fragments depend only on (tn, k) -> hoist once per wave, reuse over tm.
    const int tnA = wave & 3;
    const int nA  = tnA * 16 + l15;
    const int tm0 = wave >> 2;          // 0 or 1; tm = tm0 + 2*i, i = 0..2 (tm<5)

    // ======== layer 1 aggregation: P = Â(80x76) · h0(76x64) =================
    {
        v2f bf[KSTEPS];
        #pragma unroll
        for (int k = 0; k < KSTEPS; ++k) {
            bf[k].x = sH[(4 * k + 2 * lhalf + 0) * HID + nA];
            bf[k].y = sH[(4 * k + 2 * lhalf + 1) * HID + nA];
        }
        v8f acc[3];
        #pragma unroll
        for (int i = 0; i < 3; ++i) {
            int tm = tm0 + 2 * i;
            if (tm >= 5) break;                     // wave-uniform
            int m = tm * 16 + l15;
            v8f c = {};
            #pragma unroll
            for (int k = 0; k < KSTEPS; ++k) {
                v2f a = *(const v2f*)&sA[m * JP + 4 * k + 2 * lhalf];
                c = __builtin_amdgcn_wmma_f32_16x16x4_f32(
                        false, a, false, bf[k], (short)0, c, false, false);
            }
            acc[i] = c;
        }
        __syncthreads();
        float bias = sB1[nA];
        #pragma unroll
        for (int i = 0; i < 3; ++i) {
            int tm = tm0 + 2 * i;
            if (tm >= 5) break;
            int mbase = tm * 16 + 8 * lhalf;        // C layout: row = r + 8*lhalf
            #pragma unroll
            for (int r = 0; r < 8; ++r)
                sH[(mbase + r) * HID + nA] = gelu_exact(acc[i][r] + bias);
        }
        __syncthreads();
    }

    // ======== G = Â · h1 (same tiling), raw result back into sH =============
    {
        v2f bf[KSTEPS];
        #pragma unroll
        for (int k = 0; k < KSTEPS; ++k) {
            bf[k].x = sH[(4 * k + 2 * lhalf + 0) * HID + nA];
            bf[k].y = sH[(4 * k + 2 * lhalf + 1) * HID + nA];
        }
        v8f acc[3];
        #pragma unroll
        for (int i = 0; i < 3; ++i) {
            int tm = tm0 + 2 * i;
            if (tm >= 5) break;
            int m = tm * 16 + l15;
            v8f c = {};
            #pragma unroll
            for (int k = 0; k < KSTEPS; ++k) {
                v2f a = *(const v2f*)&sA[m * JP + 4 * k + 2 * lhalf];
                c = __builtin_amdgcn_wmma_f32_16x16x4_f32(
                        false, a, false, bf[k], (short)0, c, false, false);
            }
            acc[i] = c;
        }
        __syncthreads();
        #pragma unroll
        for (int i = 0; i < 3; ++i) {
            int tm = tm0 + 2 * i;
            if (tm >= 5) break;
            int mbase = tm * 16 + 8 * lhalf;
            #pragma unroll
            for (int r = 0; r < 8; ++r)
                sH[(mbase + r) * HID + nA] = acc[i][r];
        }
        __syncthreads();
    }

    // ======== out = gelu(G · W2 + b2): 5x16 tiles, K=64 =====================
    // Wave w owns column blocks tn = w and w+8. Hoist the 16 W2 B-fragments
    // per tn once (global/L2, 32 VGPRs), sweep 5 row tiles from LDS.
    float* outg = out + (size_t)g * (J_N * OUT_D);
    #pragma unroll
    for (int half = 0; half < 2; ++half) {
        int tn = wave + 8 * half;
        int n  = tn * 16 + l15;
        v2f bf[16];
        #pragma unroll
        for (int k = 0; k < 16; ++k) {
            bf[k].x = W2[(4 * k + 2 * lhalf + 0) * OUT_D + n];
            bf[k].y = W2[(4 * k + 2 * lhalf + 1) * OUT_D + n];
        }
        float bias = sB2[n];
        #pragma unroll
        for (int tm = 0; tm < 5; ++tm) {
            int m = tm * 16 + l15;
            v8f c = {};
            #pragma unroll
            for (int k = 0; k < 16; ++k) {
                v2f a = *(const v2f*)&sH[m * HID + 4 * k + 2 * lhalf];
                c = __builtin_amdgcn_wmma_f32_16x16x4_f32(
                        false, a, false, bf[k], (short)0, c, false, false);
            }
            int mbase = tm * 16 + 8 * lhalf;
            #pragma unroll
            for (int r = 0; r < 8; ++r) {
                int mr = mbase + r;
                if (mr < J_N)
                    outg[mr * OUT_D + n] = gelu_exact(c[r] + bias);
            }
        }
    }
}

// ---------------------------------------------------------------------------
extern "C" void kernel_launch(void* const* d_in, const int* in_sizes, int n_in,
                              void* d_out, int out_size, void* d_ws, size_t ws_size,
                              hipStream_t stream) {
    const float* x  = (const float*)d_in[0];
    const int*   ei = (const int*)d_in[1];     // edge_index, (2, 300) flat
    const float* W1 = (const float*)d_in[2];
    const float* b1 = (const float*)d_in[3];
    const float* W2 = (const float*)d_in[4];
    const float* b2 = (const float*)d_in[5];
    float* out = (float*)d_out;
    float* Adj = (float*)d_ws;                 // JP*JP floats = 25.6 KB

    build_adj_kernel<<<1, 256, 0, stream>>>(ei, Adj);
    gcn_fused_kernel<<<BT, 256, 0, stream>>>(x, W1, b1, W2, b2, Adj, out);
}